// TopKActivation_86904368268017
// MI455X (gfx1250) — compile-verified
//
#include <hip/hip_runtime.h>
#include <stdint.h>

#define ROWS 8192
#define COLS 4096
#define TOPK 64
#define NT   256   // 8 wave32s per workgroup

typedef __attribute__((ext_vector_type(4))) float    v4f;
typedef __attribute__((ext_vector_type(4))) uint32_t v4u;

// Order-preserving float-bits -> uint32 key: larger key <=> larger float.
__device__ __forceinline__ uint32_t bits2key(uint32_t u) {
  return (u & 0x80000000u) ? ~u : (u | 0x80000000u);
}
__device__ __forceinline__ float key2f(uint32_t k) {
  uint32_t u = (k & 0x80000000u) ? (k ^ 0x80000000u) : ~k;
  return __uint_as_float(u);
}

// Inclusive suffix sum across a wave32: returns sum of v over lanes [lane..31].
__device__ __forceinline__ uint32_t wave_suffix_scan(uint32_t v, int lane) {
#pragma unroll
  for (int off = 1; off < 32; off <<= 1) {
    const uint32_t t = __shfl_down(v, off, 32);
    if (lane + off < 32) v += t;
  }
  return v;
}

// Digit selection over 256 bins with only 2 workgroup barriers.
// Precondition: histogram complete (barrier passed). Reads own bin, pre-clears
// it for the next pass, wave-scans, combines across the 8 waves, and the unique
// crossing thread publishes s_sel[0]=digit, s_sel[1]=rank remaining in digit.
// Also re-zeros the compaction/tie counters behind the final barrier.
__device__ __forceinline__ void suffix_select(uint32_t* s_hist, uint32_t* s_wtot,
                                              uint32_t* s_sel, uint32_t* s_cnt,
                                              uint32_t* s_eq, uint32_t kk, int tid) {
  const int lane = tid & 31;
  const int wv   = tid >> 5;

  const uint32_t h = s_hist[tid];
  s_hist[tid] = 0u;                          // fold clear for next pass
  const uint32_t wsuf = wave_suffix_scan(h, lane);
  if (lane == 0) s_wtot[wv] = wsuf;          // wave total (bins wv*32..wv*32+31)
  if (tid == 0) { *s_cnt = 0u; *s_eq = 0u; }
  __syncthreads();

  uint32_t add_above = 0u;
#pragma unroll
  for (int w = 0; w < 8; ++w)
    add_above += (w > wv) ? s_wtot[w] : 0u;

  const uint32_t suffix = wsuf + add_above;  // sum hist[tid..255]
  const uint32_t above  = suffix - h;        // sum hist[tid+1..255]
  if (above < kk && kk <= suffix) {
    s_sel[0] = (uint32_t)tid;
    s_sel[1] = kk - above;
  }
  __syncthreads();
}

__global__ void __launch_bounds__(NT)
topk_scatter_kernel(const float* __restrict__ x, float* __restrict__ out) {
  __shared__ uint32_t s_keys[COLS];   // 16 KB: row as monotonic keys
  __shared__ uint32_t s_lA[COLS];     // 16 KB: candidate list (ping)
  __shared__ uint32_t s_lB[COLS];     // 16 KB: candidate list (pong)
  __shared__ uint32_t s_hist[256];
  __shared__ uint32_t s_wtot[8];
  __shared__ uint32_t s_sel[2];
  __shared__ uint32_t s_cnt;
  __shared__ uint32_t s_eq;

  const int tid = threadIdx.x;
  const int row = blockIdx.x;

  // ---- Stage row into LDS via CDNA5 async global->LDS B128 copies (GVS) ----
  const uint64_t sbase    = (uint64_t)(uintptr_t)x;
  const uint32_t base_off = (uint32_t)row * (uint32_t)(COLS * 4);
#pragma unroll
  for (int it = 0; it < (COLS * 4) / (NT * 16); ++it) {      // 4 iterations
    const int      elem = (it * NT + tid) * 4;
    const uint32_t voff = base_off + (uint32_t)elem * 4u;
    const uint32_t lds  = (uint32_t)(uintptr_t)(&s_keys[elem]);
    asm volatile("global_load_async_to_lds_b128 %0, %1, %2"
                 :
                 : "v"(lds), "v"(voff), "s"(sbase)
                 : "memory");
  }
  asm volatile("s_wait_asynccnt 0" ::: "memory");

  // In-place bits->key conversion: each lane converts exactly the chunks it
  // async-loaded, so only the per-wave asynccnt wait is needed before this.
  {
    v4u* p = (v4u*)s_keys;
#pragma unroll
    for (int i = tid; i < COLS / 4; i += NT) {
      v4u v = p[i];
      v.x = bits2key(v.x); v.y = bits2key(v.y);
      v.z = bits2key(v.z); v.w = bits2key(v.w);
      p[i] = v;
    }
  }
  s_hist[tid] = 0u;                 // independent of key data; pre-barrier OK
  __syncthreads();

  uint32_t kk = TOPK;
  uint32_t T, d, n;
  const v4u* kp = (const v4u*)s_keys;

  // ---- Pass 1: bits [31:24] over all 4096 keys (vectorized LDS reads) ----
  for (int i = tid; i < COLS / 4; i += NT) {
    const v4u v = kp[i];
    atomicAdd(&s_hist[v.x >> 24], 1u);
    atomicAdd(&s_hist[v.y >> 24], 1u);
    atomicAdd(&s_hist[v.z >> 24], 1u);
    atomicAdd(&s_hist[v.w >> 24], 1u);
  }
  __syncthreads();
  suffix_select(s_hist, s_wtot, s_sel, &s_cnt, &s_eq, kk, tid);
  d  = s_sel[0];
  kk = s_sel[1];
  T  = d << 24;
  for (int i = tid; i < COLS / 4; i += NT) {             // compact -> listA
    const v4u v = kp[i];
    if ((v.x >> 24) == d) s_lA[atomicAdd(&s_cnt, 1u)] = v.x;
    if ((v.y >> 24) == d) s_lA[atomicAdd(&s_cnt, 1u)] = v.y;
    if ((v.z >> 24) == d) s_lA[atomicAdd(&s_cnt, 1u)] = v.z;
    if ((v.w >> 24) == d) s_lA[atomicAdd(&s_cnt, 1u)] = v.w;
  }
  __syncthreads();
  n = s_cnt;

  // ---- Pass 2: bits [23:16] over candidate list A ----
  for (int i = tid; i < (int)n; i += NT)
    atomicAdd(&s_hist[(s_lA[i] >> 16) & 255u], 1u);
  __syncthreads();
  suffix_select(s_hist, s_wtot, s_sel, &s_cnt, &s_eq, kk, tid);
  d  = s_sel[0];
  kk = s_sel[1];
  T |= d << 16;
  for (int i = tid; i < (int)n; i += NT) {               // compact A -> B
    const uint32_t key = s_lA[i];
    if (((key >> 16) & 255u) == d) s_lB[atomicAdd(&s_cnt, 1u)] = key;
  }
  __syncthreads();
  n = s_cnt;

  // ---- Pass 3: bits [15:8] over candidate list B ----
  for (int i = tid; i < (int)n; i += NT)
    atomicAdd(&s_hist[(s_lB[i] >> 8) & 255u], 1u);
  __syncthreads();
  suffix_select(s_hist, s_wtot, s_sel, &s_cnt, &s_eq, kk, tid);
  d  = s_sel[0];
  kk = s_sel[1];
  T |= d << 8;
  for (int i = tid; i < (int)n; i += NT) {               // compact B -> A
    const uint32_t key = s_lB[i];
    if (((key >> 8) & 255u) == d) s_lA[atomicAdd(&s_cnt, 1u)] = key;
  }
  __syncthreads();
  n = s_cnt;

  // ---- Pass 4: bits [7:0] over candidate list A ----
  for (int i = tid; i < (int)n; i += NT)
    atomicAdd(&s_hist[s_lA[i] & 255u], 1u);
  __syncthreads();
  suffix_select(s_hist, s_wtot, s_sel, &s_cnt, &s_eq, kk, tid);
  d  = s_sel[0];
  kk = s_sel[1];                    // #ties at threshold key to keep (>=1)
  T |= d;

  const uint32_t eq_budget = kk;

  // ---- Scatter: coalesced non-temporal B128 stores, full row written ----
  float* __restrict__ orow = out + (size_t)row * (size_t)COLS;
  for (int i = tid; i < COLS / 4; i += NT) {
    const v4u kv = kp[i];
    uint32_t k4[4] = {kv.x, kv.y, kv.z, kv.w};
    float    r[4];
#pragma unroll
    for (int j = 0; j < 4; ++j) {
      const uint32_t key = k4[j];
      float o = 0.0f;
      if (key > T) {
        o = key2f(key);
      } else if (key == T) {
        if (atomicAdd(&s_eq, 1u) < eq_budget) o = key2f(key);
      }
      r[j] = o;
    }
    v4f v = {r[0], r[1], r[2], r[3]};
    __builtin_nontemporal_store(v, (v4f*)(orow + i * 4));
  }
}

extern "C" void kernel_launch(void* const* d_in, const int* in_sizes, int n_in,
                              void* d_out, int out_size, void* d_ws, size_t ws_size,
                              hipStream_t stream) {
  const float* x   = (const float*)d_in[0];
  float*       out = (float*)d_out;
  topk_scatter_kernel<<<dim3(ROWS), dim3(NT), 0, stream>>>(x, out);
}